// SarvamMLAMoE_20392504721497
// MI455X (gfx1250) — compile-verified
//
#include <hip/hip_runtime.h>
#include <hip/hip_bf16.h>
#include <math.h>
#include <stdint.h>

#define T_TOK 2048
#define HID 1024
#define NEXP 16
#define TOPK 4
#define NGRP 4
#define INTER 512
#define SINTER 1024
#define RSCALE 2.5f
#define KC 128   // K chunk staged in LDS

typedef __attribute__((ext_vector_type(2))) float v2f;
typedef __attribute__((ext_vector_type(8))) float v8f;

// exact pointee type from the builtin's declaration
typedef int v4i_t __attribute__((__vector_size__(4 * sizeof(int))));
typedef __attribute__((address_space(1))) v4i_t gv4i_t;
typedef __attribute__((address_space(3))) v4i_t lv4i_t;

#if defined(__has_builtin)
#if __has_builtin(__builtin_amdgcn_global_load_async_to_lds_b128)
#define USE_ASYNC_LDS 1
#endif
#endif

__device__ __forceinline__ void async_fence() {
#if defined(USE_ASYNC_LDS)
#if __has_builtin(__builtin_amdgcn_s_wait_asynccnt)
    __builtin_amdgcn_s_wait_asynccnt(0);
#else
    asm volatile("s_wait_asynccnt 0x0" ::: "memory");
#endif
#endif
}

// copy 8 consecutive floats global -> LDS (per calling thread)
__device__ __forceinline__ void stage8(const float* src, float* dst) {
#if defined(USE_ASYNC_LDS)
    __builtin_amdgcn_global_load_async_to_lds_b128(
        (gv4i_t*)(uintptr_t)(src),     (lv4i_t*)(uintptr_t)(dst),     0, 0);
    __builtin_amdgcn_global_load_async_to_lds_b128(
        (gv4i_t*)(uintptr_t)(src + 4), (lv4i_t*)(uintptr_t)(dst + 4), 0, 0);
#else
    *(float4*)(dst)     = *(const float4*)(src);
    *(float4*)(dst + 4) = *(const float4*)(src + 4);
#endif
}

__device__ __forceinline__ v8f wmma_f32(v2f a, v2f b, v8f c) {
    return __builtin_amdgcn_wmma_f32_16x16x4_f32(false, a, false, b, (short)0, c, false, false);
}

__device__ __forceinline__ float silu_f(float g) {
    return g / (1.0f + __expf(-g));
}

// ---------------------------------------------------------------------------
__global__ void zero_counts_kernel(int* counts) {
    if (threadIdx.x < NEXP) counts[threadIdx.x] = 0;
}

// ---------------------------------------------------------------------------
// Router: fp32 logits -> sigmoid -> grouped top-k -> per-expert token lists
// ---------------------------------------------------------------------------
__global__ __launch_bounds__(256) void router_kernel(
    const float* __restrict__ x,       // [T][H]
    const float* __restrict__ gate_w,  // [E][H]
    const float* __restrict__ e_bias,  // [E]
    int* __restrict__ counts,          // [E]
    int* __restrict__ toklist,         // [E][T]
    float* __restrict__ wlist)         // [E][T]
{
    int t = blockIdx.x * blockDim.x + threadIdx.x;
    if (t >= T_TOK) return;

    const float* xr = x + (size_t)t * HID;
    float s[NEXP], sb[NEXP];

    for (int e = 0; e < NEXP; ++e) {
        const float* gw = gate_w + (size_t)e * HID;
        float acc = 0.0f;
        for (int k = 0; k < HID; k += 4) {
            float4 xv = *(const float4*)(xr + k);
            float4 gv = *(const float4*)(gw + k);
            acc += xv.x * gv.x + xv.y * gv.y + xv.z * gv.z + xv.w * gv.w;
        }
        float sc = 1.0f / (1.0f + __expf(-acc));
        s[e]  = sc;
        sb[e] = sc + e_bias[e];
    }

    float gs[NGRP];
    for (int g = 0; g < NGRP; ++g) {
        float m1 = -3.4e38f, m2 = -3.4e38f;
        for (int j = 0; j < NEXP / NGRP; ++j) {
            float v = sb[g * (NEXP / NGRP) + j];
            if (v > m1) { m2 = m1; m1 = v; }
            else if (v > m2) m2 = v;
        }
        gs[g] = m1 + m2;
    }
    int g1 = 0;
    for (int g = 1; g < NGRP; ++g) if (gs[g] > gs[g1]) g1 = g;
    int g2 = -1; float g2v = -3.4e38f;
    for (int g = 0; g < NGRP; ++g) {
        if (g == g1) continue;
        if (gs[g] > g2v) { g2v = gs[g]; g2 = g; }
    }

    bool allow[NEXP], taken[NEXP];
    for (int e = 0; e < NEXP; ++e) {
        int g = e / (NEXP / NGRP);
        allow[e] = (g == g1) || (g == g2);
        taken[e] = false;
    }

    int   ids[TOPK];
    float wv[TOPK];
    float wsum = 0.0f;
    for (int kp = 0; kp < TOPK; ++kp) {
        int best = -1; float bv = -3.4e38f;
        for (int e = 0; e < NEXP; ++e) {
            if (!allow[e] || taken[e]) continue;
            if (sb[e] > bv) { bv = sb[e]; best = e; }
        }
        taken[best] = true;
        ids[kp] = best;
        wv[kp]  = s[best];
        wsum   += s[best];
    }
    float scale = RSCALE / wsum;

    for (int kp = 0; kp < TOPK; ++kp) {
        int e = ids[kp];
        int slot = atomicAdd(&counts[e], 1);
        toklist[(size_t)e * T_TOK + slot] = t;
        wlist[(size_t)e * T_TOK + slot]   = wv[kp] * scale;
    }
}

// ---------------------------------------------------------------------------
// Shared expert, M=16 token tiles (128 workgroups), direct store to out.
// ---------------------------------------------------------------------------
__global__ __launch_bounds__(256) void shared_expert_kernel(
    const float* __restrict__ x,    // [T][H]
    const float* __restrict__ swg,  // [H][SI]
    const float* __restrict__ swu,  // [H][SI]
    const float* __restrict__ swd,  // [SI][H]
    float* __restrict__ out)        // [T][H]
{
    __shared__ float Xs[16][KC + 4];   // stride%64==4 -> conflict-free
    __shared__ float Hs[16][512 + 4];

    const int tid  = threadIdx.x;
    const int lane = tid & 31;
    const int wv   = tid >> 5;
    const int m    = lane & 15;
    const int n    = lane & 15;
    const int kh   = (lane >> 4) << 1;
    const int rhi  = (lane >> 4) << 3;

    const int tbase = blockIdx.x * 16;

    v8f accD[8];
    #pragma unroll
    for (int i = 0; i < 8; ++i) accD[i] = (v8f){};

    for (int ic = 0; ic < SINTER; ic += 512) {
        v8f accG[4], accU[4];
        #pragma unroll
        for (int i = 0; i < 4; ++i) { accG[i] = (v8f){}; accU[i] = (v8f){}; }

        for (int k0 = 0; k0 < HID; k0 += KC) {
            {   // stage X chunk 16x128 (8 floats / thread)
                int r = tid >> 4;
                int c = (tid & 15) * 8;
                stage8(x + (size_t)(tbase + r) * HID + k0 + c, &Xs[r][c]);
                async_fence();
            }
            __syncthreads();

            for (int kk = 0; kk < KC; kk += 4) {
                v2f a;
                a.x = Xs[m][kk + kh];
                a.y = Xs[m][kk + kh + 1];
                int kg = k0 + kk + kh;
                const float* g0 = swg + (size_t)kg * SINTER + ic + wv * 64 + n;
                const float* u0 = swu + (size_t)kg * SINTER + ic + wv * 64 + n;
                #pragma unroll
                for (int tI = 0; tI < 4; ++tI) {
                    v2f bg; bg.x = g0[tI * 16]; bg.y = g0[SINTER + tI * 16];
                    accG[tI] = wmma_f32(a, bg, accG[tI]);
                    v2f bu; bu.x = u0[tI * 16]; bu.y = u0[SINTER + tI * 16];
                    accU[tI] = wmma_f32(a, bu, accU[tI]);
                }
            }
            __syncthreads();
        }

        #pragma unroll
        for (int tI = 0; tI < 4; ++tI) {
            #pragma unroll
            for (int r = 0; r < 8; ++r) {
                Hs[r + rhi][wv * 64 + tI * 16 + n] = silu_f(accG[tI][r]) * accU[tI][r];
            }
        }
        __syncthreads();

        for (int kk = 0; kk < 512; kk += 4) {
            v2f a;
            a.x = Hs[m][kk + kh];
            a.y = Hs[m][kk + kh + 1];
            const float* d0 = swd + (size_t)(ic + kk + kh) * HID + wv * 128 + n;
            #pragma unroll
            for (int tI = 0; tI < 8; ++tI) {
                v2f b; b.x = d0[tI * 16]; b.y = d0[HID + tI * 16];
                accD[tI] = wmma_f32(a, b, accD[tI]);
            }
        }
        __syncthreads();
    }

    #pragma unroll
    for (int r = 0; r < 8; ++r) {
        float* dst = out + (size_t)(tbase + r + rhi) * HID + wv * 128 + n;
        #pragma unroll
        for (int tI = 0; tI < 8; ++tI) dst[tI * 16] = accD[tI][r];
    }
}

// ---------------------------------------------------------------------------
// Routed experts, M=32 token tiles: each B-fragment feeds TWO wmma (2 row
// tiles) -> 16 FLOP per L2 byte on the weight stream. grid = (expert, tile)
// ---------------------------------------------------------------------------
__global__ __launch_bounds__(256) void moe_routed_kernel(
    const float* __restrict__ x,       // [T][H]
    const float* __restrict__ w_gate,  // [E][H][I]
    const float* __restrict__ w_up,    // [E][H][I]
    const float* __restrict__ w_down,  // [E][I][H]
    const int*   __restrict__ counts,  // [E]
    const int*   __restrict__ toklist, // [E][T]
    const float* __restrict__ wlist,   // [E][T]
    float* __restrict__ out)           // [T][H]
{
    const int e    = blockIdx.x;
    const int tile = blockIdx.y;
    const int cnt  = counts[e];
    if (tile * 32 >= cnt) return;      // uniform -> EXEC all-1s for WMMA

    __shared__ float Xs[32][KC + 4];
    __shared__ float Hs[32][INTER + 4];
    __shared__ int   toks[32];
    __shared__ float wts[32];

    const int tid  = threadIdx.x;
    const int lane = tid & 31;
    const int wv   = tid >> 5;
    const int m    = lane & 15;
    const int n    = lane & 15;
    const int kh   = (lane >> 4) << 1;
    const int rhi  = (lane >> 4) << 3;

    if (tid < 32) {
        int s = tile * 32 + tid;
        if (s < cnt) {
            toks[tid] = toklist[(size_t)e * T_TOK + s];
            wts[tid]  = wlist[(size_t)e * T_TOK + s];
        } else {
            toks[tid] = 0;
            wts[tid]  = 0.0f;
        }
    }
    __syncthreads();

    const float* Wg = w_gate + (size_t)e * HID * INTER;
    const float* Wu = w_up   + (size_t)e * HID * INTER;
    const float* Wd = w_down + (size_t)e * INTER * HID;

    v8f accG[2][4], accU[2][4];
    #pragma unroll
    for (int rt = 0; rt < 2; ++rt)
        #pragma unroll
        for (int i = 0; i < 4; ++i) { accG[rt][i] = (v8f){}; accU[rt][i] = (v8f){}; }

    for (int k0 = 0; k0 < HID; k0 += KC) {
        {   // gather X chunk 32x128 (16 floats / thread)
            int r = tid >> 3;
            int c = (tid & 7) * 16;
            const float* src = x + (size_t)toks[r] * HID + k0 + c;
            stage8(src,     &Xs[r][c]);
            stage8(src + 8, &Xs[r][c + 8]);
            async_fence();
        }
        __syncthreads();

        for (int kk = 0; kk < KC; kk += 4) {
            v2f a0, a1;
            a0.x = Xs[m][kk + kh];      a0.y = Xs[m][kk + kh + 1];
            a1.x = Xs[m + 16][kk + kh]; a1.y = Xs[m + 16][kk + kh + 1];
            int kg = k0 + kk + kh;
            const float* g0 = Wg + (size_t)kg * INTER + wv * 64 + n;
            const float* u0 = Wu + (size_t)kg * INTER + wv * 64 + n;
            #pragma unroll
            for (int tI = 0; tI < 4; ++tI) {
                v2f bg; bg.x = g0[tI * 16]; bg.y = g0[INTER + tI * 16];
                accG[0][tI] = wmma_f32(a0, bg, accG[0][tI]);
                accG[1][tI] = wmma_f32(a1, bg, accG[1][tI]);
                v2f bu; bu.x = u0[tI * 16]; bu.y = u0[INTER + tI * 16];
                accU[0][tI] = wmma_f32(a0, bu, accU[0][tI]);
                accU[1][tI] = wmma_f32(a1, bu, accU[1][tI]);
            }
        }
        __syncthreads();
    }

    // H = silu(G) * U -> LDS (both row tiles)
    #pragma unroll
    for (int rt = 0; rt < 2; ++rt) {
        #pragma unroll
        for (int tI = 0; tI < 4; ++tI) {
            #pragma unroll
            for (int r = 0; r < 8; ++r) {
                Hs[r + rhi + rt * 16][wv * 64 + tI * 16 + n] =
                    silu_f(accG[rt][tI][r]) * accU[rt][tI][r];
            }
        }
    }
    __syncthreads();

    // down-proj: D[32x1024] = H @ Wd, wave owns 128 cols, B shared by 2 tiles
    v8f accD[2][8];
    #pragma unroll
    for (int rt = 0; rt < 2; ++rt)
        #pragma unroll
        for (int i = 0; i < 8; ++i) accD[rt][i] = (v8f){};

    for (int kk = 0; kk < INTER; kk += 4) {
        v2f a0, a1;
        a0.x = Hs[m][kk + kh];      a0.y = Hs[m][kk + kh + 1];
        a1.x = Hs[m + 16][kk + kh]; a1.y = Hs[m + 16][kk + kh + 1];
        const float* d0 = Wd + (size_t)(kk + kh) * HID + wv * 128 + n;
        #pragma unroll
        for (int tI = 0; tI < 8; ++tI) {
            v2f b; b.x = d0[tI * 16]; b.y = d0[HID + tI * 16];
            accD[0][tI] = wmma_f32(a0, b, accD[0][tI]);
            accD[1][tI] = wmma_f32(a1, b, accD[1][tI]);
        }
    }

    #pragma unroll
    for (int rt = 0; rt < 2; ++rt) {
        #pragma unroll
        for (int r = 0; r < 8; ++r) {
            int   row = r + rhi + rt * 16;
            int   tok = toks[row];
            float w   = wts[row];
            float* dst = out + (size_t)tok * HID + wv * 128 + n;
            #pragma unroll
            for (int tI = 0; tI < 8; ++tI)
                atomicAdd(dst + tI * 16, accD[rt][tI][r] * w);
        }
    }
}

// ---------------------------------------------------------------------------
extern "C" void kernel_launch(void* const* d_in, const int* in_sizes, int n_in,
                              void* d_out, int out_size, void* d_ws, size_t ws_size,
                              hipStream_t stream) {
    const float* x       = (const float*)d_in[0];
    const float* gate_w  = (const float*)d_in[1];
    const float* e_bias  = (const float*)d_in[2];
    const float* w_gate  = (const float*)d_in[3];
    const float* w_up    = (const float*)d_in[4];
    const float* w_down  = (const float*)d_in[5];
    const float* sw_gate = (const float*)d_in[6];
    const float* sw_up   = (const float*)d_in[7];
    const float* sw_down = (const float*)d_in[8];
    float* out = (float*)d_out;

    int*   counts  = (int*)d_ws;
    int*   toklist = counts + 16;                      // [16][2048] ints
    float* wlist   = (float*)(toklist + NEXP * T_TOK); // [16][2048] floats

    zero_counts_kernel<<<1, 32, 0, stream>>>(counts);
    router_kernel<<<T_TOK / 256, 256, 0, stream>>>(x, gate_w, e_bias,
                                                   counts, toklist, wlist);
    shared_expert_kernel<<<T_TOK / 16, 256, 0, stream>>>(x, sw_gate, sw_up,
                                                         sw_down, out);
    dim3 grid(NEXP, T_TOK / 32);
    moe_routed_kernel<<<grid, 256, 0, stream>>>(x, w_gate, w_up, w_down,
                                                counts, toklist, wlist, out);
}